// SITHModule_61607010894467
// MI455X (gfx1250) — compile-verified
//
#include <hip/hip_runtime.h>
#include <hip/hip_bf16.h>

// ---------------------------------------------------------------------------
// SITH module: diagonal-linear scan (t' = A*t + B*DT*x) + projection L @ t'.
// Strategy:
//   init   : compute cA[100], cB[100]=B*DT, A_S[100]=A^64, Lpad[16][100] on device
//   pass1a : per (chunk k, feature f): local chunk end-state from zero start
//   pass1b : sequential prefix over 64 chunks (in place) -> chunk start states
//   pass2  : per (chunk, 16-feature tile) wave: replay 64 steps; projection via
//            V_WMMA_F32_16X16X4_F32 (fp32 WMMA, 25 K-steps covering tau=100)
// ---------------------------------------------------------------------------

typedef __attribute__((ext_vector_type(2))) float v2f;
typedef __attribute__((ext_vector_type(8))) float v8f;

#define NTAU   100
#define SEQ    4096
#define NF     1024
#define NCH    64          // chunks
#define CS     64          // chunk size (NCH*CS == SEQ)
#define NROWS  12          // L rows after slicing
#define KSTEPS 25          // 100 / 4

// ws float layout
#define WS_CA   0          // [100]
#define WS_CB   100        // [100]  (B * DT folded)
#define WS_AS   200        // [100]  (A^CS)
#define WS_L    320        // [16*100] padded L
#define WS_C    2048       // [NCH][NTAU][NF] chunk states (26.2 MB)

// ---------------- device-side constant generation (fp64, matches numpy) ----

__device__ __forceinline__ double s_of(int i) {
  return 4.0 / (0.1 * pow(1.1, (double)i));   // K / tau_star
}
__device__ double Dval(int i, int j) {
  if (i < 1 || i > NTAU - 2) return 0.0;
  double inv = 1.0 / (s_of(i + 1) - s_of(i - 1));
  if (j == i + 1) return inv;
  if (j == i - 1) return -inv;
  return 0.0;
}
__device__ double D2val(int i, int j) {
  double r = 0.0;
  for (int k = i - 1; k <= i + 1; k += 2)
    if (k >= 0 && k <= NTAU - 1) r += Dval(i, k) * Dval(k, j);
  return r;
}
__device__ double D4val(int i, int j) {
  double r = 0.0;
  for (int k = i - 2; k <= i + 2; k += 2)
    if (k >= 0 && k <= NTAU - 1) r += D2val(i, k) * D2val(k, j);
  return r;
}

__global__ void sith_init(float* __restrict__ ws) {
  int t = threadIdx.x;
  const double DTd = 1.0 / 300.0;
  if (t < NTAU) {
    double s  = s_of(t);
    double a  = 1.0 - s * DTd;           // ALPHA = 1
    double A1 = pow(a, 10.0);            // n1 = round(DUR/DT)   = 10
    double B1 = (1.0 - A1) / s;
    double A2 = pow(a, 9.0);             // n2 = round(DELAY/DT) = 9
    double A  = A1 * A2;
    double B  = A2 * B1;
    ws[WS_CA + t] = (float)A;
    ws[WS_CB + t] = (float)(B * DTd);    // fold item = x*DT into B
    double Af = (double)(float)A;        // chunk-combine uses fp32 A^CS
    ws[WS_AS + t] = (float)pow(Af, (double)CS);
  }
  for (int idx = t; idx < 16 * NTAU; idx += blockDim.x) {
    int m = idx / NTAU, j = idx % NTAU;
    float val = 0.f;
    if (m < NROWS) {
      int i = 4 + 8 * m;                 // rows 4..95 step 8 (K..NTAU-K, ::8)
      double si = s_of(i);
      val = (float)(pow(si, 5.0) / 24.0 * D4val(i, j));  // (-1)^4 = +1, 4! = 24
    }
    ws[WS_L + idx] = val;
  }
}

// ---------------- pass 1a: local chunk end-states from zero ----------------

__global__ void sith_pass1a(const float* __restrict__ x, float* __restrict__ ws) {
  __shared__ float sA[NTAU], sB[NTAU];
  int t = threadIdx.x;
  if (t < NTAU) { sA[t] = ws[WS_CA + t]; sB[t] = ws[WS_CB + t]; }
  __syncthreads();
  int gid = blockIdx.x * blockDim.x + t;      // 0 .. NCH*NF-1
  int f = gid & (NF - 1);
  int k = gid >> 10;
  const float* xp = x + (size_t)k * CS * NF + f;
  float st[NTAU];
#pragma unroll
  for (int i = 0; i < NTAU; ++i) st[i] = 0.f;
  for (int n = 0; n < CS; ++n) {
    float xv = xp[(size_t)n * NF];
#pragma unroll
    for (int i = 0; i < NTAU; ++i) st[i] = fmaf(sA[i], st[i], sB[i] * xv);
  }
  float* c = ws + WS_C + (size_t)k * NTAU * NF + f;
#pragma unroll
  for (int i = 0; i < NTAU; ++i) c[(size_t)i * NF] = st[i];
}

// ---------------- pass 1b: prefix over chunks (in place) -------------------

__global__ void sith_pass1b(float* __restrict__ ws) {
  int gid = blockIdx.x * blockDim.x + threadIdx.x;   // (i, f)
  if (gid >= NTAU * NF) return;
  int i = gid >> 10;
  float As = ws[WS_AS + i];
  float* c = ws + WS_C + gid;
  float tv = 0.f;
  for (int k = 0; k < NCH; ++k) {
    float lc = c[(size_t)k * NTAU * NF];
    c[(size_t)k * NTAU * NF] = tv;                   // becomes chunk START state
    tv = fmaf(As, tv, lc);
  }
}

// ---------------- pass 2: replay + fp32 WMMA projection --------------------
// wave32 owns one 16-feature tile. Fragment mapping (documented 32-bit WMMA
// layouts): lane l -> col n16 = l%16, half h = l/16; A/B VGPR v holds
// K = 4*kk + v + 2*h; C/D VGPR r holds row M = r + 8*h.

__global__ void __launch_bounds__(256)
sith_pass2(const float* __restrict__ x, const float* __restrict__ ws,
           float* __restrict__ out) {
  __shared__ v2f cf[2][2 * KSTEPS];   // (A_i, B_i) per half / per (kk,v)
  int tid = threadIdx.x;
  if (tid < 2 * 2 * KSTEPS) {
    int h = tid / (2 * KSTEPS), j = tid % (2 * KSTEPS);
    int i = 4 * (j >> 1) + (j & 1) + 2 * h;          // tau index for this slot
    v2f ab; ab.x = ws[WS_CA + i]; ab.y = ws[WS_CB + i];
    cf[h][j] = ab;
  }
  __syncthreads();

  int wave = tid >> 5;
  int lane = tid & 31;
  int h    = lane >> 4;
  int n16  = lane & 15;
  int k      = blockIdx.x >> 3;                      // chunk
  int fgroup = blockIdx.x & 7;
  int f = (fgroup * 8 + wave) * 16 + n16;

  // L fragments (A-matrix), padded rows 12..15 are zero in ws.
  v2f fl[KSTEPS];
  const float* Lp = ws + WS_L + n16 * NTAU;
#pragma unroll
  for (int kk = 0; kk < KSTEPS; ++kk) {
    int kb = 4 * kk + 2 * h;
    v2f a; a.x = Lp[kb]; a.y = Lp[kb + 1];
    fl[kk] = a;
  }

  // chunk start state (B-matrix fragments).
  v2f st[KSTEPS];
  const float* ts = ws + WS_C + (size_t)k * NTAU * NF + f;
#pragma unroll
  for (int kk = 0; kk < KSTEPS; ++kk) {
    int i0 = 4 * kk + 2 * h;
    v2f s; s.x = ts[(size_t)i0 * NF]; s.y = ts[(size_t)(i0 + 1) * NF];
    st[kk] = s;
  }

  const float* xp = x + (size_t)k * CS * NF + f;
  float* op = out + (size_t)k * CS * NROWS * NF + f;

  for (int n = 0; n < CS; ++n) {
    float xv = xp[(size_t)n * NF];
#pragma unroll
    for (int kk = 0; kk < KSTEPS; ++kk) {
      v2f a0 = cf[h][2 * kk], a1 = cf[h][2 * kk + 1];
      st[kk].x = fmaf(a0.x, st[kk].x, a0.y * xv);
      st[kk].y = fmaf(a1.x, st[kk].y, a1.y * xv);
    }
    v8f acc = {};
#pragma unroll
    for (int kk = 0; kk < KSTEPS; ++kk) {
      acc = __builtin_amdgcn_wmma_f32_16x16x4_f32(
          false, fl[kk], false, st[kk], (short)0, acc, false, false);
    }
#pragma unroll
    for (int r = 0; r < 8; ++r) {
      int M = 8 * h + r;                 // C/D row for this half
      if (M < NROWS) op[((size_t)n * NROWS + M) * NF] = acc[r];
    }
  }
}

// ---------------------------------------------------------------------------

extern "C" void kernel_launch(void* const* d_in, const int* in_sizes, int n_in,
                              void* d_out, int out_size, void* d_ws, size_t ws_size,
                              hipStream_t stream) {
  const float* x = (const float*)d_in[0];        // (4096, 1024) fp32
  float* ws  = (float*)d_ws;
  float* out = (float*)d_out;                    // (4096, 12, 1024) fp32

  sith_init  <<<1, 256, 0, stream>>>(ws);
  sith_pass1a<<<(NCH * NF) / 256, 256, 0, stream>>>(x, ws);
  sith_pass1b<<<(NTAU * NF + 255) / 256, 256, 0, stream>>>(ws);
  sith_pass2 <<<NCH * 8, 256, 0, stream>>>(x, ws, out);
}